// Sponge_82059645157798
// MI455X (gfx1250) — compile-verified
//
#include <hip/hip_runtime.h>
#include <math.h>

#ifndef __has_builtin
#define __has_builtin(x) 0
#endif

// ---------------- problem constants (from reference) ----------------
#define NT           256          // threads per block (8 wave32)
#define ROWS         2            // batch rows per workgroup
#define INPUT_SIZE   1024
#define SPONGE_SIZE  1024
#define ACT_SIZE     256
#define RECALL_SIZE  256
#define STORE_SIZE   512
#define DEPTH        16
#define BF_DEPTH     10
#define IN_BF_DEPTH  10
#define MEM_SIZE     (DEPTH * STORE_SIZE)        /* 8192 */
#define N_OUT_MEM    (DEPTH * ACT_SIZE)          /* 4096 */
#define OUT_BF_SIZE  (N_OUT_MEM + SPONGE_SIZE)   /* 5120 */
#define OUT_BF_DEPTH 13
#define OUTPUT_SIZE  1024
#define BATCH        8192

// ---------------- cos/sin table layout in d_ws (float2 entries) ----
#define CS_IN_N    (IN_BF_DEPTH * (INPUT_SIZE / 2))          /* 5120   */
#define CS_SP_N    (DEPTH * BF_DEPTH * (SPONGE_SIZE / 2))    /* 81920  */
#define CS_OUT_N   (OUT_BF_DEPTH * (OUT_BF_SIZE / 2))        /* 33280  */
#define CS_IN_OFF  0
#define CS_SP_OFF  (CS_IN_OFF + CS_IN_N)
#define CS_OUT_OFF (CS_SP_OFF + CS_SP_N)
#define CS_TOTAL   (CS_OUT_OFF + CS_OUT_N)                   /* 120320 */

typedef unsigned int u32x4 __attribute__((ext_vector_type(4)));
typedef int          i32x4 __attribute__((ext_vector_type(4)));
typedef int          i32x8 __attribute__((ext_vector_type(8)));

// ---------------- TDM helpers (1-D tile, 4-byte elements) -----------
__device__ __forceinline__ void tdm_load_1d(const void* gsrc, const void* lds_dst,
                                            unsigned nelem) {
#if __has_builtin(__builtin_amdgcn_tensor_load_to_lds)
    unsigned long long ga = (unsigned long long)gsrc;
    unsigned int lds = (unsigned int)(unsigned long long)lds_dst; // low32 = LDS offset
    u32x4 g0 = { 1u, lds, (unsigned int)ga,
                 (unsigned int)(ga >> 32) | (2u << 30) };         // type=2
    i32x8 g1 = { (int)(2u << 16),        // data_size = 4B
                 (int)(nelem << 16),     // tensor_dim0
                 (int)(1u << 16),        // tensor_dim1 = 1
                 (int)(nelem << 16),     // tile_dim0
                 0,                      // tile_dim1/2 unused
                 (int)nelem,             // dim0 stride
                 0, 0 };
    i32x4 g2 = { 0, 0, 0, 0 };
    i32x4 g3 = { 0, 0, 0, 0 };
#if defined(__clang_major__) && (__clang_major__ >= 23)
    i32x8 g4 = { 0, 0, 0, 0, 0, 0, 0, 0 };
    __builtin_amdgcn_tensor_load_to_lds(g0, g1, g2, g3, g4, 0);
#else
    __builtin_amdgcn_tensor_load_to_lds(g0, g1, g2, g3, 0);
#endif
#endif
}

__device__ __forceinline__ void tdm_store_1d(void* gdst, const void* lds_src,
                                             unsigned nelem) {
#if __has_builtin(__builtin_amdgcn_tensor_store_from_lds)
    unsigned long long ga = (unsigned long long)gdst;
    unsigned int lds = (unsigned int)(unsigned long long)lds_src;
    u32x4 g0 = { 1u, lds, (unsigned int)ga,
                 (unsigned int)(ga >> 32) | (2u << 30) };
    i32x8 g1 = { (int)(2u << 16), (int)(nelem << 16), (int)(1u << 16),
                 (int)(nelem << 16), 0, (int)nelem, 0, 0 };
    i32x4 g2 = { 0, 0, 0, 0 };
    i32x4 g3 = { 0, 0, 0, 0 };
#if defined(__clang_major__) && (__clang_major__ >= 23)
    i32x8 g4 = { 0, 0, 0, 0, 0, 0, 0, 0 };
    __builtin_amdgcn_tensor_store_from_lds(g0, g1, g2, g3, g4, 0);
#else
    __builtin_amdgcn_tensor_store_from_lds(g0, g1, g2, g3, 0);
#endif
#endif
}

__device__ __forceinline__ void tdm_wait0() {
#if __has_builtin(__builtin_amdgcn_s_wait_tensorcnt)
    __builtin_amdgcn_s_wait_tensorcnt(0);
#else
    asm volatile("s_wait_tensorcnt 0" ::: "memory");
#endif
}

// ---------------- kernel 0: precompute cos/sin tables ----------------
__global__ __launch_bounds__(256) void cs_precompute(
    const float* __restrict__ in_bf, const float* __restrict__ sp_bf,
    const float* __restrict__ out_bf, float2* __restrict__ cs)
{
    int i = blockIdx.x * blockDim.x + threadIdx.x;
    if (i >= CS_TOTAL) return;
    float a;
    if (i < CS_SP_OFF)        a = in_bf[i - CS_IN_OFF];
    else if (i < CS_OUT_OFF)  a = sp_bf[i - CS_SP_OFF];
    else                      a = out_bf[i - CS_OUT_OFF];
    float s, c;
    sincosf(a, &s, &c);
    cs[i] = make_float2(c, s);
}

// ---------------- ping-pong butterfly over ROWS LDS buffers ----------
// Pair j (j in [0,E)) reads src[j/2 + (j%2)*E] and +E/2; writes dst[j], dst[j+E].
// Thread handles pair-pairs (2j',2j'+1): 4 conflict-free b32 LDS reads,
// two b64 LDS writes, one 16B coefficient load (software-pipelined).
// One barrier per layer. Returns the buffer holding the result.
template <int NPP, bool USE_CS>
__device__ __forceinline__ float* butterfly_pp(
    float* src, float* dst, int rowStride,
    const float* __restrict__ tab, int nlayers, int t, int E)
{
    const int Eh = E >> 1;
    const size_t lstride = (size_t)(USE_CS ? 2 * E : E);
    float4 q_cur[NPP], q_nxt[NPP];
#pragma unroll
    for (int p = 0; p < NPP; ++p) {
        const int jp = t + p * NT;
        if (USE_CS) {
            q_cur[p] = *reinterpret_cast<const float4*>(tab + 4 * (size_t)jp);
        } else {
            float2 a = *reinterpret_cast<const float2*>(tab + 2 * (size_t)jp);
            q_cur[p].x = a.x; q_cur[p].y = a.y; q_cur[p].z = 0.f; q_cur[p].w = 0.f;
        }
    }
    for (int l = 0; l < nlayers; ++l) {
        float a0[ROWS][NPP], a1[ROWS][NPP], a2v[ROWS][NPP], a3[ROWS][NPP];
#pragma unroll
        for (int p = 0; p < NPP; ++p) {
            const int jp = t + p * NT;
#pragma unroll
            for (int r = 0; r < ROWS; ++r) {
                const float* b = src + r * rowStride;
                a0[r][p]  = b[jp];
                a1[r][p]  = b[jp + Eh];
                a2v[r][p] = b[jp + E];
                a3[r][p]  = b[jp + E + Eh];
            }
        }
        // software-pipelined coefficient load for the next layer
        if (l + 1 < nlayers) {
            const float* tn = tab + (size_t)(l + 1) * lstride;
#pragma unroll
            for (int p = 0; p < NPP; ++p) {
                const int jp = t + p * NT;
                if (USE_CS) {
                    q_nxt[p] = *reinterpret_cast<const float4*>(tn + 4 * (size_t)jp);
                } else {
                    float2 a = *reinterpret_cast<const float2*>(tn + 2 * (size_t)jp);
                    q_nxt[p].x = a.x; q_nxt[p].y = a.y; q_nxt[p].z = 0.f; q_nxt[p].w = 0.f;
                }
            }
        }
#pragma unroll
        for (int p = 0; p < NPP; ++p) {
            const int jp = t + p * NT;
            float c0, s0, c1, s1;
            if (USE_CS) { c0 = q_cur[p].x; s0 = q_cur[p].y; c1 = q_cur[p].z; s1 = q_cur[p].w; }
            else { sincosf(q_cur[p].x, &s0, &c0); sincosf(q_cur[p].y, &s1, &c1); }
#pragma unroll
            for (int r = 0; r < ROWS; ++r) {
                float* d = dst + r * rowStride;
                float2 lo, hi;
                lo.x = fmaf(c0, a0[r][p],  s0 * a1[r][p]);
                hi.x = fmaf(c0, a1[r][p], -s0 * a0[r][p]);
                lo.y = fmaf(c1, a2v[r][p],  s1 * a3[r][p]);
                hi.y = fmaf(c1, a3[r][p], -s1 * a2v[r][p]);
                *reinterpret_cast<float2*>(d + 2 * jp)     = lo;
                *reinterpret_cast<float2*>(d + 2 * jp + E) = hi;
            }
        }
        __syncthreads();
        if (l + 1 < nlayers) {
#pragma unroll
            for (int p = 0; p < NPP; ++p) q_cur[p] = q_nxt[p];
        }
        float* tmp = src; src = dst; dst = tmp;
    }
    return src;   // buffer that holds the result of the last layer
}

// ---------------- main kernel: ROWS rows per workgroup ---------------
template <bool USE_CS>
__global__ __launch_bounds__(NT) void sponge_main(
    const float* __restrict__ X, const float* __restrict__ scales,
    const float* __restrict__ act_bias, const int* __restrict__ recall_idx,
    const int* __restrict__ out_mem_idx,
    const float* __restrict__ tab_in, const float* __restrict__ tab_sp,
    const float* __restrict__ tab_out, float* __restrict__ out)
{
    // single LDS arena with dead-region reuse for ping-pong buffers
    __shared__ float smem[ROWS * (MEM_SIZE + SPONGE_SIZE + OUT_BF_SIZE)]; // 114688 B
    float* s_mem    = smem;                           // ROWS x 8192
    float* s_sponge = smem + ROWS * MEM_SIZE;         // ROWS x 1024
    float* s_out    = s_sponge + ROWS * SPONGE_SIZE;  // ROWS x 5120
    float* sp_alt   = s_out;   // stage-time alternate sponge (s_out dead till gather)
    float* out_alt  = s_mem;   // output-time alternate buffer (mem dead after gather)

    const int t    = threadIdx.x;
    const int row0 = blockIdx.x * ROWS;
    const float* Xr = X + (size_t)row0 * INPUT_SIZE;

    // ---- stage X rows (ROWS*4KB, contiguous) into LDS via TDM -------
#if __has_builtin(__builtin_amdgcn_tensor_load_to_lds)
    if ((t >> 5) == 0) {                 // wave 0 only (uniform per-wave branch)
        tdm_load_1d(Xr, s_sponge, ROWS * INPUT_SIZE);
        tdm_wait0();
    }
    __syncthreads();
    for (int k = t; k < ROWS * INPUT_SIZE; k += NT)
        s_sponge[k] *= scales[k & (INPUT_SIZE - 1)];
#else
    for (int k = t; k < ROWS * INPUT_SIZE; k += NT)
        s_sponge[k] = Xr[k] * scales[k & (INPUT_SIZE - 1)];
#endif
    __syncthreads();

    // ---- input butterfly: W=1024, 10 layers (even -> ends in s_sponge)
    float* sp = butterfly_pp<1, USE_CS>(s_sponge, sp_alt, SPONGE_SIZE, tab_in,
                                        IN_BF_DEPTH, t, INPUT_SIZE / 2);

    // ---- 16 sponge stages -------------------------------------------
    const size_t sp_stage = (size_t)BF_DEPTH * (SPONGE_SIZE / 2) * (USE_CS ? 2 : 1);
    for (int i = 0; i < DEPTH; ++i) {
        if (USE_CS && t == 0 && (i + 1) < DEPTH)
            __builtin_prefetch(tab_sp + (size_t)(i + 1) * sp_stage, 0, 1);

        float* alt = (sp == s_sponge) ? sp_alt : s_sponge;
        sp = butterfly_pp<1, USE_CS>(sp, alt, SPONGE_SIZE,
                                     tab_sp + (size_t)i * sp_stage,
                                     BF_DEPTH, t, SPONGE_SIZE / 2);

        // append sponge[0:512] to mem; capture act input (sponge[512+t])
        float actin[ROWS];
#pragma unroll
        for (int r = 0; r < ROWS; ++r) {
            const float* spr = sp + r * SPONGE_SIZE;
            actin[r] = spr[STORE_SIZE + t];
            s_mem[r * MEM_SIZE + i * STORE_SIZE + t]       = spr[t];
            s_mem[r * MEM_SIZE + i * STORE_SIZE + t + 256] = spr[t + 256];
        }
        __syncthreads();  // appends visible; all pre-overwrite reads done

        const int   ri   = recall_idx[i * RECALL_SIZE + t];
        const float bias = act_bias[i * ACT_SIZE + t];
#pragma unroll
        for (int r = 0; r < ROWS; ++r) {
            float* spr = sp + r * SPONGE_SIZE;
            float rv = s_mem[r * MEM_SIZE + ri];
            float x  = actin[r] + bias;
            float sq = sqrtf(fmaf(x, x, 4.0f));   // c = 2/NEUTRAL_CURVATURE
            spr[t]                       = rv;                      // recall
            spr[RECALL_SIZE + 2 * t]     = 0.5f * ( x + sq - 2.0f); // act(+x)
            spr[RECALL_SIZE + 2 * t + 1] = 0.5f * (-x + sq - 2.0f); // act(-x)
            // spr[768:1024] unchanged in place
        }
        __syncthreads();
    }

    // ---- build pre = [mem[out_mem_idx] | sponge] into s_out ---------
    for (int k = t; k < N_OUT_MEM; k += NT) {
        int mi = out_mem_idx[k];
#pragma unroll
        for (int r = 0; r < ROWS; ++r)
            s_out[r * OUT_BF_SIZE + k] = s_mem[r * MEM_SIZE + mi];
    }
    for (int k = t; k < SPONGE_SIZE; k += NT) {
#pragma unroll
        for (int r = 0; r < ROWS; ++r)
            s_out[r * OUT_BF_SIZE + N_OUT_MEM + k] = sp[r * SPONGE_SIZE + k];
    }
    __syncthreads();

    // ---- output butterfly: W=5120, 13 layers (odd -> ends in out_alt)
    float* fin = butterfly_pp<5, USE_CS>(s_out, out_alt, OUT_BF_SIZE, tab_out,
                                         OUT_BF_DEPTH, t, OUT_BF_SIZE / 2);

    // ---- write out[:, :1024] ----------------------------------------
#if __has_builtin(__builtin_amdgcn_tensor_store_from_lds)
    if ((t >> 5) == 0) {                 // wave 0 issues DMA stores
#pragma unroll
        for (int r = 0; r < ROWS; ++r)
            tdm_store_1d(out + (size_t)(row0 + r) * OUTPUT_SIZE,
                         fin + r * OUT_BF_SIZE, OUTPUT_SIZE);
        // no explicit wait: S_ENDPGM performs an implicit wait-idle,
        // and LDS lives until every wave of the workgroup has ended.
    }
#else
#pragma unroll
    for (int r = 0; r < ROWS; ++r) {
        float4*       dst = reinterpret_cast<float4*>(out + (size_t)(row0 + r) * OUTPUT_SIZE);
        const float4* src = reinterpret_cast<const float4*>(fin + r * OUT_BF_SIZE);
        dst[t] = src[t];   // 256 threads x 16B = 4KB row
    }
#endif
}

// ---------------- host-side launch ----------------------------------
extern "C" void kernel_launch(void* const* d_in, const int* in_sizes, int n_in,
                              void* d_out, int out_size, void* d_ws, size_t ws_size,
                              hipStream_t stream)
{
    const float* X      = (const float*)d_in[0];
    const float* scales = (const float*)d_in[1];
    const float* in_bf  = (const float*)d_in[2];
    const float* sp_bf  = (const float*)d_in[3];
    const float* actb   = (const float*)d_in[4];
    const float* out_bf = (const float*)d_in[5];
    const int*   recall = (const int*)d_in[6];
    const int*   outmem = (const int*)d_in[7];
    float*       out    = (float*)d_out;
    (void)in_sizes; (void)n_in; (void)out_size;

    const size_t cs_bytes = (size_t)CS_TOTAL * sizeof(float2);
    if (d_ws && ws_size >= cs_bytes) {
        float2* cs = (float2*)d_ws;
        cs_precompute<<<(CS_TOTAL + 255) / 256, 256, 0, stream>>>(in_bf, sp_bf, out_bf, cs);
        const float* base = (const float*)cs;
        sponge_main<true><<<BATCH / ROWS, NT, 0, stream>>>(
            X, scales, actb, recall, outmem,
            base + 2 * (size_t)CS_IN_OFF,
            base + 2 * (size_t)CS_SP_OFF,
            base + 2 * (size_t)CS_OUT_OFF, out);
    } else {
        // workspace too small: compute sin/cos in-kernel from raw angles
        sponge_main<false><<<BATCH / ROWS, NT, 0, stream>>>(
            X, scales, actb, recall, outmem, in_bf, sp_bf, out_bf, out);
    }
}